// _MultiOrganismLinear_80315888435838
// MI455X (gfx1250) — compile-verified
//
#include <hip/hip_runtime.h>

typedef __attribute__((ext_vector_type(2))) float v2f;
typedef __attribute__((ext_vector_type(8))) float v8f;

namespace {
constexpr int kB   = 8;
constexpr int kS   = 4096;
constexpr int kD   = 1536;
constexpr int kOut = 2048;

constexpr int BM  = 128;
constexpr int BN  = 256;
constexpr int BK  = 32;
constexpr int LDX = BK + 4;   // 36-float row stride for X tile (conflict-free A frags)
constexpr int LDW = BN + 8;   // 264-float row stride for W tile (2*264 mod 64 = 16 -> disjoint halves)
constexpr int NK  = kD / BK;  // 48 K-slices
constexpr int XSZ = BM * LDX; // 4608 floats / buffer
constexpr int WSZ = BK * LDW; // 8448 floats / buffer
}

// Async direct global->LDS copy (gfx1250, ASYNCcnt-tracked), GVS addressing:
// mem_addr = SGPR64 base + VGPR32 offset; VDST VGPR holds LDS byte address.
__device__ __forceinline__ void async_b128(unsigned lds_byte, unsigned voff_byte,
                                           unsigned long long sbase) {
    asm volatile("global_load_async_to_lds_b128 %0, %1, %2 offset:0"
                 :
                 : "v"(lds_byte), "v"(voff_byte), "s"(sbase)
                 : "memory");
}
__device__ __forceinline__ void wait_async0() {
    asm volatile("s_wait_asynccnt 0" ::: "memory");
}

__global__ __launch_bounds__(256, 1)
void mol_wmma_f32(const float* __restrict__ x,
                  const int* __restrict__ org_idx,
                  const float* __restrict__ w,
                  const float* __restrict__ bias,
                  float* __restrict__ out)
{
    __shared__ float Xs[2 * XSZ];   //  36864 B (double buffered)
    __shared__ float Ws[2 * WSZ];   //  67584 B (double buffered)

    const int tid  = threadIdx.x;
    const int lane = tid & 31;
    const int wid  = tid >> 5;
    const int half = lane >> 4;      // 16-lane half of the wave
    const int m16  = lane & 15;

    const int batch  = blockIdx.z;
    const int Mblock = blockIdx.y * BM;
    const int Nblock = blockIdx.x * BN;

    const int org = org_idx[batch];

    const int waveM = (wid >> 2) * 64;   // 0 or 64
    const int waveN = (wid & 3) * 64;    // 0,64,128,192

    const float* xblk = x + ((size_t)batch * kS + Mblock) * kD;
    const float* wblk = w + (size_t)org * kD * kOut + Nblock;

    // LDS aperture addresses keep the byte offset in addr[31:0]
    const unsigned xs0 = (unsigned)(uintptr_t)&Xs[0];
    const unsigned ws0 = (unsigned)(uintptr_t)&Ws[0];

    // Per-thread copy slices
    const int xrow = tid >> 3, xc = (tid & 7) * 4;   // X: 128 rows x 8 float4 (4 steps of 32 rows)
    const int wc   = (tid & 63) * 4;                 // W: 32 rows x 64 float4 (8 steps of 4 rows)
    const int wrow = tid >> 6;

    auto async_tile = [&](int kt, int buf) {
        const unsigned long long xbase =
            (unsigned long long)(uintptr_t)(xblk + kt * BK);
        const unsigned long long wbase =
            (unsigned long long)(uintptr_t)(wblk + (size_t)kt * BK * kOut);
#pragma unroll
        for (int i = 0; i < 4; ++i) {
            const int xr = xrow + i * 32;
            async_b128(xs0 + (unsigned)((buf * XSZ + xr * LDX + xc) * 4),
                       (unsigned)((xr * kD + xc) * 4), xbase);
        }
#pragma unroll
        for (int i = 0; i < 8; ++i) {
            const int wr = wrow + i * 4;
            async_b128(ws0 + (unsigned)((buf * WSZ + wr * LDW + wc) * 4),
                       (unsigned)((wr * kOut + wc) * 4), wbase);
        }
    };

    v8f acc[4][4];
    const v8f vzero = {0.f, 0.f, 0.f, 0.f, 0.f, 0.f, 0.f, 0.f};
#pragma unroll
    for (int mi = 0; mi < 4; ++mi)
#pragma unroll
        for (int ni = 0; ni < 4; ++ni)
            acc[mi][ni] = vzero;

    async_tile(0, 0);
    wait_async0();
    __syncthreads();

    for (int kt = 0; kt < NK; ++kt) {
        const int buf = kt & 1;
        if (kt + 1 < NK) async_tile(kt + 1, buf ^ 1);  // slice-depth prefetch into LDS

        const float* Xb = &Xs[buf * XSZ];
        const float* Wb = &Ws[buf * WSZ];
#pragma unroll
        for (int ks = 0; ks < BK; ks += 4) {
            const int kk = ks + 2 * half;  // ISA A/B lane-half K offset
            v2f a[4], b[4];
#pragma unroll
            for (int mi = 0; mi < 4; ++mi) {
                const float* p = &Xb[(waveM + mi * 16 + m16) * LDX + kk];
                v2f av; av.x = p[0]; av.y = p[1];   // ds_load_b64, conflict-free
                a[mi] = av;
            }
#pragma unroll
            for (int ni = 0; ni < 4; ++ni) {
                const float* p = &Wb[kk * LDW + waveN + ni * 16 + m16];
                v2f bv; bv.x = p[0]; bv.y = p[LDW];
                b[ni] = bv;
            }
#pragma unroll
            for (int mi = 0; mi < 4; ++mi)
#pragma unroll
                for (int ni = 0; ni < 4; ++ni)
                    acc[mi][ni] = __builtin_amdgcn_wmma_f32_16x16x4_f32(
                        false, a[mi], false, b[ni], (short)0, acc[mi][ni],
                        false, false);
        }

        wait_async0();       // my async writes into buf^1 have landed in LDS
        __syncthreads();     // all waves done: safe to read buf^1 / overwrite buf
    }

    // Epilogue: C/D layout — VGPR r holds M=r (lanes 0-15) and M=r+8 (lanes 16-31)
#pragma unroll
    for (int ni = 0; ni < 4; ++ni) {
        const int col = Nblock + waveN + ni * 16 + m16;
        const float bv = bias[(size_t)org * kOut + col];
#pragma unroll
        for (int mi = 0; mi < 4; ++mi) {
            const int mrow0 = Mblock + waveM + mi * 16 + 8 * half;
            v8f c = acc[mi][ni];
#pragma unroll
            for (int r = 0; r < 8; ++r) {
                out[((size_t)batch * kS + mrow0 + r) * (size_t)kOut + col] = c[r] + bv;
            }
        }
    }
}

extern "C" void kernel_launch(void* const* d_in, const int* in_sizes, int n_in,
                              void* d_out, int out_size, void* d_ws, size_t ws_size,
                              hipStream_t stream) {
    const float* x   = (const float*)d_in[0];
    const int*   idx = (const int*)d_in[1];
    const float* w   = (const float*)d_in[2];
    const float* b   = (const float*)d_in[3];
    float*       o   = (float*)d_out;
    (void)in_sizes; (void)n_in; (void)out_size; (void)d_ws; (void)ws_size;

    dim3 grid(kOut / BN, kS / BM, kB);   // 8 x 32 x 8 = 2048 workgroups
    mol_wmma_f32<<<grid, dim3(256), 0, stream>>>(x, idx, w, b, o);
}